// BinarizeLayer_22754736734796
// MI455X (gfx1250) — compile-verified
//
#include <hip/hip_runtime.h>

// ---------------------------------------------------------------------------
// BinarizeLayer: out[b,f] = (med[f] > 0 && in[b,f] >= med[f]) ? 1.0f : 0.0f
//
// Memory-bound: 268 MB of streaming traffic -> ~11.5 us at 23.3 TB/s.
// Strategy (MI455X / gfx1250):
//   * b128 (float4) per-lane loads/stores -> 512 B contiguous per wave32 op
//   * non-temporal hints on the streamed input/output (268 MB > 192 MB L2),
//     keeping the 16 KB medians vector cache-resident
//   * column-strip mapping: medians float4 loaded once per thread, then the
//     thread streams ROWS_PER_BLOCK rows; no integer div/mod anywhere
// ---------------------------------------------------------------------------

typedef float v4f __attribute__((ext_vector_type(4)));

#define THREADS_X      256
#define ROWS_PER_BLOCK 8

__global__ __launch_bounds__(THREADS_X)
void binarize_vec4_kernel(const float* __restrict__ in,
                          const float* __restrict__ med,
                          float* __restrict__ out,
                          int fvec,    // features / 4
                          int rows)    // batch
{
    const int c4 = blockIdx.x * THREADS_X + threadIdx.x;
    if (c4 >= fvec) return;

    const v4f* __restrict__ in4  = (const v4f*)in;
    const v4f* __restrict__ med4 = (const v4f*)med;
    v4f*       __restrict__ out4 = (v4f*)out;

    // Medians: tiny (16 KB), reused by every row -> regular temporal load,
    // held in VGPRs for the whole strip.
    const v4f m = med4[c4];

    const int row0 = blockIdx.y * ROWS_PER_BLOCK;

#pragma unroll
    for (int r = 0; r < ROWS_PER_BLOCK; ++r) {
        const int row = row0 + r;
        if (row >= rows) break;
        const size_t idx = (size_t)row * (size_t)fvec + (size_t)c4;

        // Streamed exactly once -> non-temporal (TH=NT) so it doesn't evict
        // the medians vector from cache.
        const v4f x = __builtin_nontemporal_load(&in4[idx]);

        v4f o;
        o.x = (m.x > 0.0f && x.x >= m.x) ? 1.0f : 0.0f;
        o.y = (m.y > 0.0f && x.y >= m.y) ? 1.0f : 0.0f;
        o.z = (m.z > 0.0f && x.z >= m.z) ? 1.0f : 0.0f;
        o.w = (m.w > 0.0f && x.w >= m.w) ? 1.0f : 0.0f;

        __builtin_nontemporal_store(o, &out4[idx]);
    }
}

// Scalar fallback for feature counts not divisible by 4 (not hit with the
// reference shapes, but keeps kernel_launch total).
__global__ __launch_bounds__(THREADS_X)
void binarize_scalar_kernel(const float* __restrict__ in,
                            const float* __restrict__ med,
                            float* __restrict__ out,
                            long long n, int features)
{
    long long i = (long long)blockIdx.x * THREADS_X + threadIdx.x;
    const long long stride = (long long)gridDim.x * THREADS_X;
    for (; i < n; i += stride) {
        const int f = (int)(i % features);
        const float m = med[f];
        const float x = in[i];
        out[i] = (m > 0.0f && x >= m) ? 1.0f : 0.0f;
    }
}

extern "C" void kernel_launch(void* const* d_in, const int* in_sizes, int n_in,
                              void* d_out, int out_size, void* d_ws, size_t ws_size,
                              hipStream_t stream)
{
    (void)n_in; (void)d_ws; (void)ws_size; (void)out_size;

    const float* in  = (const float*)d_in[0];
    const float* med = (const float*)d_in[1];
    float*       out = (float*)d_out;

    const long long n        = (long long)in_sizes[0]; // 8192*4096
    const int       features = in_sizes[1];            // 4096
    const int       rows     = (int)(n / (long long)features);

    if ((features & 3) == 0) {
        const int fvec = features >> 2;                     // 1024
        const int gx   = (fvec + THREADS_X - 1) / THREADS_X; // 4
        const int gy   = (rows + ROWS_PER_BLOCK - 1) / ROWS_PER_BLOCK; // 1024
        binarize_vec4_kernel<<<dim3(gx, gy), dim3(THREADS_X), 0, stream>>>(
            in, med, out, fvec, rows);
    } else {
        const long long nthreads = (n + 0) ;
        int blocks = (int)((nthreads + THREADS_X - 1) / THREADS_X);
        if (blocks > 65535) blocks = 65535;
        binarize_scalar_kernel<<<dim3(blocks), dim3(THREADS_X), 0, stream>>>(
            in, med, out, n, features);
    }
}